// ParagraphGAT_7988639171401
// MI455X (gfx1250) — compile-verified
//
#include <hip/hip_runtime.h>
#include <math.h>

// ---------------------------------------------------------------------------
// GATv2 (2 layers) for MI455X / gfx1250, wave32.
// GEMMs use V_WMMA_F32_16X16X4_F32 (exact fp32, matrix pipe), with
// compile-time column counts so all inner-loop loads use immediate offsets.
// Edge phases are bandwidth-bound; use native f32 atomics.
// ---------------------------------------------------------------------------

typedef float v2f __attribute__((ext_vector_type(2)));
typedef float v8f __attribute__((ext_vector_type(8)));

#define DF 64          // feature dim (fixed by the model)
#define NEG_SLOPE 0.2f

// ---------------- utility: fill ----------------
__global__ void fill_f32_kernel(float* __restrict__ p, float v, size_t n) {
  size_t i = (size_t)blockIdx.x * blockDim.x + threadIdx.x;
  if (i < n) p[i] = v;
}

// ---------------- fp32 WMMA GEMM: out[n_rows,KC] = A[n_rows,64] @ W[64,KC]
// KC compile-time; one wave computes NT consecutive 16x16 output tiles,
// reusing the A fragment across NT WMMAs. All k-loop loads use immediate
// offsets (no per-step address math). n_rows % 16 == 0.
template <int KC, int NT>
__global__ void gemm_x64_wmma(const float* __restrict__ A,
                              const float* __restrict__ W,
                              float* __restrict__ out, int n_rows) {
  constexpr int TILES_N = KC / 16;
  constexpr int GROUPS  = TILES_N / NT;     // tile-groups per row strip
  int wid  = blockIdx.x * (blockDim.x >> 5) + (threadIdx.x >> 5);
  int lane = threadIdx.x & 31;
  int total = (n_rows >> 4) * GROUPS;
  if (wid >= total) return;
  int tm = wid / GROUPS;
  int tg = wid % GROUPS;
  int half = lane >> 4;   // 0 -> K pair {0,1}; 1 -> K pair {2,3}  (ISA 16x4 f32 A layout)
  int l16  = lane & 15;   // A row within tile / B,C column within tile

  const float* arow = A + (size_t)(tm * 16 + l16) * DF + 2 * half;
  const float* b0[NT];
#pragma unroll
  for (int t = 0; t < NT; ++t)
    b0[t] = W + (2 * half) * KC + (tg * NT + t) * 16 + l16;

  v8f c[NT];
#pragma unroll
  for (int t = 0; t < NT; ++t) c[t] = (v8f){};

#pragma unroll
  for (int kk = 0; kk < DF; kk += 4) {
    v2f a = *(const v2f*)(arow + kk);            // A[m][kk+2h], A[m][kk+2h+1]
#pragma unroll
    for (int t = 0; t < NT; ++t) {
      v2f b;
      b.x = b0[t][kk * KC];                      // W[kk+2h  ][col]  (imm offset)
      b.y = b0[t][kk * KC + KC];                 // W[kk+2h+1][col]  (imm offset)
      c[t] = __builtin_amdgcn_wmma_f32_16x16x4_f32(false, a, false, b,
                                                   (short)0, c[t], false, false);
    }
  }
  // C layout: VGPR i -> row tm*16 + i + 8*half, col (tg*NT+t)*16 + l16
#pragma unroll
  for (int t = 0; t < NT; ++t) {
    float* o = out + (size_t)(tm * 16 + 8 * half) * KC + (tg * NT + t) * 16 + l16;
#pragma unroll
    for (int i = 0; i < 8; ++i) o[i * KC] = c[t][i];
  }
}

// ---------------- float atomic max via int/uint atomics ----------------
__device__ inline void atomicMaxFloat(float* addr, float v) {
  if (v >= 0.0f) atomicMax((int*)addr, __float_as_int(v));
  else           atomicMin((unsigned int*)addr, __float_as_uint(v));
}

// ---------------- edge pass 1: GATv2 logits + segment max ----------------
// one wave per edge; lanes cover d = lane and lane+32.
template <int H>
__global__ void edge_logits_kernel(const float* __restrict__ xl,
                                   const float* __restrict__ xr,
                                   const float* __restrict__ att,
                                   const int* __restrict__ src,
                                   const int* __restrict__ dst,
                                   int E, int N,
                                   float* __restrict__ lg,
                                   float* __restrict__ mmax) {
  int wid  = blockIdx.x * (blockDim.x >> 5) + (threadIdx.x >> 5);
  int lane = threadIdx.x & 31;
  int ET = E + N;
  if (wid >= ET) return;
  int s = (wid < E) ? src[wid] : (wid - E);   // implicit self-loops
  int t = (wid < E) ? dst[wid] : (wid - E);
  const float* pl = xl + (size_t)s * (H * DF);
  const float* pr = xr + (size_t)t * (H * DF);
#pragma unroll
  for (int h = 0; h < H; ++h) {
    float v0 = pl[h * DF + lane]      + pr[h * DF + lane];
    float v1 = pl[h * DF + lane + 32] + pr[h * DF + lane + 32];
    v0 = (v0 > 0.0f) ? v0 : v0 * NEG_SLOPE;
    v1 = (v1 > 0.0f) ? v1 : v1 * NEG_SLOPE;
    float p = v0 * att[h * DF + lane] + v1 * att[h * DF + lane + 32];
#pragma unroll
    for (int off = 16; off > 0; off >>= 1) p += __shfl_xor(p, off, 32);
    if (lane == 0) {
      lg[(size_t)wid * H + h] = p;
      atomicMaxFloat(&mmax[(size_t)t * H + h], p);
    }
  }
}

// ---------------- edge pass 2: exp(logit - max) + segment sum ----------------
template <int H>
__global__ void edge_exp_kernel(float* __restrict__ lg,
                                const float* __restrict__ mmax,
                                const int* __restrict__ dst,
                                int E, int N,
                                float* __restrict__ denom) {
  size_t i = (size_t)blockIdx.x * blockDim.x + threadIdx.x;
  size_t total = (size_t)(E + N) * H;
  if (i >= total) return;
  int e = (int)(i / H), h = (int)(i % H);
  int t = (e < E) ? dst[e] : (e - E);
  float a = __expf(lg[i] - mmax[(size_t)t * H + h]);
  lg[i] = a;                                   // in place: alpha numerator
  unsafeAtomicAdd(&denom[(size_t)t * H + h], a);
}

// ---------------- edge pass 3: scatter-add alpha * xl[src] ----------------
// one wave per (edge, head); lanes cover d = lane and lane+32.
template <int H>
__global__ void edge_scatter_kernel(const float* __restrict__ xl,
                                    const float* __restrict__ alpha,
                                    const float* __restrict__ denom,
                                    const int* __restrict__ src,
                                    const int* __restrict__ dst,
                                    int E, int N,
                                    float* __restrict__ acc) {
  int wid  = blockIdx.x * (blockDim.x >> 5) + (threadIdx.x >> 5);
  int lane = threadIdx.x & 31;
  long long total = (long long)(E + N) * H;
  if ((long long)wid >= total) return;
  int e = wid / H, h = wid % H;
  int s = (e < E) ? src[e] : (e - E);
  int t = (e < E) ? dst[e] : (e - E);
  float al = alpha[(size_t)e * H + h] / denom[(size_t)t * H + h];
  const float* pl = xl  + (size_t)s * (H * DF) + h * DF;
  float*       pa = acc + (size_t)t * (H * DF) + h * DF;
  unsafeAtomicAdd(&pa[lane],      al * pl[lane]);
  unsafeAtomicAdd(&pa[lane + 32], al * pl[lane + 32]);
}

// ---------------- finalize: head mean + bias (+ optional ReLU) ----------------
template <int H, bool RELU>
__global__ void finalize_kernel(const float* __restrict__ acc,
                                const float* __restrict__ bias,
                                float* __restrict__ out, int N) {
  int i = blockIdx.x * blockDim.x + threadIdx.x;
  if (i >= N * DF) return;
  int n = i >> 6, d = i & 63;
  float s = 0.0f;
#pragma unroll
  for (int h = 0; h < H; ++h) s += acc[(size_t)n * (H * DF) + h * DF + d];
  float v = s * (1.0f / H) + bias[d];
  if (RELU) v = fmaxf(v, 0.0f);
  out[i] = v;
}

// ---------------- one GAT layer ----------------
template <int H, bool RELU>
static void run_gat_layer(const float* xin, const float* Wl, const float* Wr,
                          const float* att, const float* bias,
                          const int* src, const int* dst, int E, int N,
                          float* xl, float* xr, float* lg, float* m, float* dn,
                          float* acc, float* out, hipStream_t stream) {
  constexpr int KC = H * DF;                 // 512 or 256
  constexpr int NT = KC / 128;               // 4 or 2 n-tiles per wave
  const int ET = E + N;
  // dense transforms (WMMA fp32)
  int waves = (N / 16) * (KC / 16 / NT);
  int gblks = (waves + 7) / 8;               // 8 waves / 256-thread block
  gemm_x64_wmma<KC, NT><<<gblks, 256, 0, stream>>>(xin, Wl, xl, N);
  gemm_x64_wmma<KC, NT><<<gblks, 256, 0, stream>>>(xin, Wr, xr, N);
  // init segment buffers
  size_t nh = (size_t)N * H, nk = (size_t)N * KC;
  fill_f32_kernel<<<(unsigned)((nh + 255) / 256), 256, 0, stream>>>(m, -INFINITY, nh);
  fill_f32_kernel<<<(unsigned)((nh + 255) / 256), 256, 0, stream>>>(dn, 0.0f, nh);
  fill_f32_kernel<<<(unsigned)((nk + 255) / 256), 256, 0, stream>>>(acc, 0.0f, nk);
  // edge phases
  edge_logits_kernel<H><<<(ET + 7) / 8, 256, 0, stream>>>(xl, xr, att, src, dst, E, N, lg, m);
  size_t eth = (size_t)ET * H;
  edge_exp_kernel<H><<<(unsigned)((eth + 255) / 256), 256, 0, stream>>>(lg, m, dst, E, N, dn);
  edge_scatter_kernel<H><<<(unsigned)((eth + 7) / 8), 256, 0, stream>>>(xl, lg, dn, src, dst, E, N, acc);
  // head mean + bias (+relu)
  finalize_kernel<H, RELU><<<((N * DF) + 255) / 256, 256, 0, stream>>>(acc, bias, out, N);
}

// ---------------- entry point ----------------
extern "C" void kernel_launch(void* const* d_in, const int* in_sizes, int n_in,
                              void* d_out, int out_size, void* d_ws, size_t ws_size,
                              hipStream_t stream) {
  const float* x    = (const float*)d_in[0];
  const int*   ei   = (const int*)d_in[1];     // [2, E]
  const float* W1l  = (const float*)d_in[2];
  const float* W1r  = (const float*)d_in[3];
  const float* att1 = (const float*)d_in[4];
  const float* b1   = (const float*)d_in[5];
  const float* W2l  = (const float*)d_in[6];
  const float* W2r  = (const float*)d_in[7];
  const float* att2 = (const float*)d_in[8];
  const float* b2   = (const float*)d_in[9];

  const int N  = in_sizes[0] / DF;      // 50000
  const int E  = in_sizes[1] / 2;       // 800000
  (void)n_in; (void)out_size; (void)ws_size;
  const int ET = E + N;

  const int* src = ei;
  const int* dst = ei + E;

  // workspace arena (floats), sized for the larger (layer-1, H=8) footprint
  float* ws = (float*)d_ws;
  const size_t K1 = (size_t)8 * DF;     // 512
  size_t o_xl  = 0;
  size_t o_xr  = o_xl  + (size_t)N * K1;
  size_t o_lg  = o_xr  + (size_t)N * K1;
  size_t o_m   = o_lg  + (size_t)ET * 8;
  size_t o_dn  = o_m   + (size_t)N * 8;
  size_t o_acc = o_dn  + (size_t)N * 8;
  size_t o_h   = o_acc + (size_t)N * K1;

  // layer 1 (H=8) -> h (ReLU'ed) in workspace
  run_gat_layer<8, true>(x, W1l, W1r, att1, b1, src, dst, E, N,
                         ws + o_xl, ws + o_xr, ws + o_lg, ws + o_m, ws + o_dn,
                         ws + o_acc, ws + o_h, stream);

  // layer 2 (H=4) -> d_out, reusing the arena (smaller footprint fits)
  run_gat_layer<4, false>(ws + o_h, W2l, W2r, att2, b2, src, dst, E, N,
                          ws + o_xl, ws + o_xr, ws + o_lg, ws + o_m, ws + o_dn,
                          ws + o_acc, (float*)d_out, stream);
}